// PINN_Acceleration_27101243638253
// MI455X (gfx1250) — compile-verified
//
#include <hip/hip_runtime.h>
#include <math.h>

// ---------------------------------------------------------------------------
// PINN acceleration: a_phys (N-body pairwise, WMMA Gram-matrix + batched
//                    trans-op pipeline + async global->LDS staging)
//                  + a_nn  (tiny 2-layer MLP, bandwidth-bound GEMVs)
//
// d2_ij = |pi|^2 + |pj|^2 - 2 pi.pj  -> pi.pj tile via V_WMMA_F32_16X16X4_F32
// a_i   = sum_j w_ij p_j - (sum_j w_ij) p_i
// ---------------------------------------------------------------------------

typedef float v2f __attribute__((ext_vector_type(2)));
typedef float v8f __attribute__((ext_vector_type(8)));

#define NB      4096
#define SOFT    0.05f
#define HDIM    64
#define KIN     (3 * NB)          // 12288
#define WG_I    128               // i rows per workgroup (8 waves x 16)
#define JSLICES 8                 // grid.y split of the j range
#define JSLICE  (NB / JSLICES)    // 512
#define JCHUNK  256               // j bodies staged in LDS at a time

// Async global->LDS B32 (ASYNCcnt path, ISA 15.18.3 op 96). The low 32 bits
// of a generic pointer into __shared__ are the LDS byte offset.
__device__ __forceinline__ void async_ld_f32(const float* g, float* lds_dst) {
  unsigned off = (unsigned)(uintptr_t)lds_dst;
  asm volatile("global_load_async_to_lds_b32 %0, %1, off"
               :: "v"(off), "v"(g) : "memory");
}
__device__ __forceinline__ void wait_async0() {
  asm volatile("s_wait_asynccnt 0x0" ::: "memory");
}

// ---------------- init: zero output and workspace ---------------------------
__global__ __launch_bounds__(256) void k_zero(float* __restrict__ out,
                                              float* __restrict__ ws) {
  int t = blockIdx.x * blockDim.x + threadIdx.x;
  if (t < KIN) out[t] = 0.0f;
  if (t < 2 * HDIM) ws[t] = 0.0f;
}

// ---------------- pairwise forces (WMMA f32 16x16x4) ------------------------
__global__ __launch_bounds__(256) void k_forces(const float* __restrict__ pos,
                                                const float* __restrict__ mass,
                                                float* __restrict__ out) {
  __shared__ float4 s_ipos[WG_I];   // x, y, z, |p|^2 of this WG's i rows
  __shared__ float  s_jx[JCHUNK];   // j chunk split by component: the async
  __shared__ float  s_jy[JCHUNK];   // gather de-strides pos[N,3] and the
  __shared__ float  s_jz[JCHUNK];   // compute-side ds_load_b32 is bank-
  __shared__ float  s_jm[JCHUNK];   // conflict-free (16 banks, broadcast).

  const int t    = threadIdx.x;
  const int lane = t & 31;
  const int wave = t >> 5;          // 0..7, one 16-row i-tile per wave
  const int r    = lane & 15;       // column / row-in-half index
  const int hi   = lane >> 4;       // which 16-lane half

  const int ibaseWG = blockIdx.x * WG_I;
  const int jbase   = blockIdx.y * JSLICE;

  // Stage this WG's i rows (pos + |p|^2) into LDS.
  if (t < WG_I) {
    int i = ibaseWG + t;
    float x = pos[3 * i + 0], y = pos[3 * i + 1], z = pos[3 * i + 2];
    s_ipos[t] = make_float4(x, y, z, x * x + y * y + z * z);
  }
  __syncthreads();

  const int iTileBase   = wave * 16;
  const int iTileGlobal = ibaseWG + iTileBase;

  // A matrix (16x4 f32, ISA 7.12.2): lane L<16 holds row M=L with K=0 in
  // VGPR0, K=1 in VGPR1; lanes 16-31 hold K=2 (z) and K=3 (pad 0).
  float4 myip = s_ipos[iTileBase + r];
  v2f A;
  A.x = hi ? myip.z : myip.x;
  A.y = hi ? 0.0f   : myip.y;

  // |p_i|^2 for the 8 C/D rows this lane owns: row m = v + 8*hi.
  float ri2v[8];
#pragma unroll
  for (int v = 0; v < 8; ++v) ri2v[v] = s_ipos[iTileBase + v + 8 * hi].w;

  float accx[8], accy[8], accz[8], accs[8];
#pragma unroll
  for (int v = 0; v < 8; ++v) { accx[v] = accy[v] = accz[v] = accs[v] = 0.0f; }

  for (int chunk = 0; chunk < JSLICE / JCHUNK; ++chunk) {
    __syncthreads();  // protect LDS reuse before re-filling
    {
      const int jb = jbase + chunk * JCHUNK;
      async_ld_f32(&pos[3 * (jb + t) + 0], &s_jx[t]);
      async_ld_f32(&pos[3 * (jb + t) + 1], &s_jy[t]);
      async_ld_f32(&pos[3 * (jb + t) + 2], &s_jz[t]);
      async_ld_f32(&mass[jb + t],          &s_jm[t]);
      wait_async0();                 // this wave's LDS writes landed
    }
    __syncthreads();                 // all waves' writes visible

    if (chunk + 1 < JSLICE / JCHUNK) {  // pull next chunk toward L2/L0
      const int jn = jbase + (chunk + 1) * JCHUNK;
      __builtin_prefetch(&pos[3 * (jn + t)], 0, 1);
      __builtin_prefetch(&mass[jn + t], 0, 1);
    }

#pragma unroll 2
    for (int jt = 0; jt < JCHUNK / 16; ++jt) {
      const int jj    = jt * 16 + r;
      const int jTile = jbase + chunk * JCHUNK + jt * 16;
      float pjx = s_jx[jj], pjy = s_jy[jj], pjz = s_jz[jj], mj = s_jm[jj];

      // B matrix (4x16 f32): VGPR0 = {K=0 (x) lanes 0-15, K=1 (y) 16-31},
      // VGPR1 = {K=2 (z) lanes 0-15, K=3 (0) 16-31}.
      v2f B;
      B.x = hi ? pjy : pjx;
      B.y = hi ? 0.0f : pjz;

      float rj2 = fmaf(pjx, pjx, fmaf(pjy, pjy, pjz * pjz));

      // Gram tile: dot[m][n] = p_i[m].p_j[n] (one matrix-pipe op / 256 pairs)
      v8f dot = {};
      dot = __builtin_amdgcn_wmma_f32_16x16x4_f32(
          false, A, false, B, (short)0, dot, false, false);

      // Phase-batched per-pair math: keeps 8 independent sqrt (then 8 rcp)
      // in flight so no TRANS32 hazard nops are needed.
      float d2a[8], ra[8], inva[8], wv[8];
#pragma unroll
      for (int v = 0; v < 8; ++v)
        d2a[v] = fmaxf(fmaf(-2.0f, dot[v], ri2v[v] + rj2), 0.0f);
#pragma unroll
      for (int v = 0; v < 8; ++v)
        ra[v] = __builtin_amdgcn_sqrtf(d2a[v]) + SOFT;
#pragma unroll
      for (int v = 0; v < 8; ++v)
        inva[v] = __builtin_amdgcn_rcpf(ra[v]);
#pragma unroll
      for (int v = 0; v < 8; ++v)
        wv[v] = mj * inva[v] * inva[v] * inva[v];

      // Diagonal masking: only the one overlapping tile per i-row needs it,
      // and there the masked lane is structural: r == v + 8*hi.
      if (iTileGlobal == jTile) {
#pragma unroll
        for (int v = 0; v < 8; ++v)
          if (r == v + 8 * hi) wv[v] = 0.0f;
      }

#pragma unroll
      for (int v = 0; v < 8; ++v) {
        accx[v] = fmaf(wv[v], pjx, accx[v]);
        accy[v] = fmaf(wv[v], pjy, accy[v]);
        accz[v] = fmaf(wv[v], pjz, accz[v]);
        accs[v] += wv[v];
      }
    }
  }

  // Reduce the j dimension across each 16-lane half (wave32; C/D halves own
  // disjoint row sets, so the reduction must stay within a half).
#pragma unroll
  for (int v = 0; v < 8; ++v) {
#pragma unroll
    for (int m = 1; m <= 8; m <<= 1) {
      accx[v] += __shfl_xor(accx[v], m, 32);
      accy[v] += __shfl_xor(accy[v], m, 32);
      accz[v] += __shfl_xor(accz[v], m, 32);
      accs[v] += __shfl_xor(accs[v], m, 32);
    }
  }

  // Writeback: lane r==v of each half owns row m = v + 8*hi.
#pragma unroll
  for (int v = 0; v < 8; ++v) {
    if (r == v) {
      int    row = v + 8 * hi;
      float4 ip  = s_ipos[iTileBase + row];
      int    ig  = ibaseWG + iTileBase + row;
      atomicAdd(&out[3 * ig + 0], accx[v] - accs[v] * ip.x);
      atomicAdd(&out[3 * ig + 1], accy[v] - accs[v] * ip.y);
      atomicAdd(&out[3 * ig + 2], accz[v] - accs[v] * ip.z);
    }
  }
}

// ---------------- MLP stage 1: hraw = x @ W1 (split-K, atomic combine) ------
__global__ __launch_bounds__(256) void k_mlp1(const float* __restrict__ x,
                                              const float* __restrict__ W1,
                                              float* __restrict__ hraw) {
  const int j    = threadIdx.x & 63;                     // output unit
  const int rep  = blockIdx.x * 4 + (threadIdx.x >> 6);  // 0..95 k-slices
  const int kPer = KIN / 96;                             // 128
  const int k0   = rep * kPer;
  float acc = 0.0f;
  for (int k = k0; k < k0 + kPer; ++k)
    acc = fmaf(x[k], W1[k * HDIM + j], acc);  // consecutive lanes -> coalesced
  atomicAdd(&hraw[j], acc);
}

// ---------------- MLP stage 2: h = tanh(hraw + b1) --------------------------
__global__ __launch_bounds__(64) void k_mlp2(const float* __restrict__ hraw,
                                             const float* __restrict__ b1,
                                             float* __restrict__ h) {
  int j = threadIdx.x;
  h[j] = tanhf(hraw[j] + b1[j]);
}

// ---------------- MLP stage 3: out += h @ W2 + b2 ---------------------------
__global__ __launch_bounds__(256) void k_mlp3(const float* __restrict__ h,
                                              const float* __restrict__ W2,
                                              const float* __restrict__ b2,
                                              float* __restrict__ out) {
  int n = blockIdx.x * blockDim.x + threadIdx.x;  // 0..12287
  float acc = b2[n];
#pragma unroll 8
  for (int j = 0; j < HDIM; ++j)
    acc = fmaf(h[j], W2[j * KIN + n], acc);       // coalesced row reads
  out[n] += acc;  // prior kernels on this stream have completed
}

// ---------------------------------------------------------------------------
extern "C" void kernel_launch(void* const* d_in, const int* in_sizes, int n_in,
                              void* d_out, int out_size, void* d_ws,
                              size_t ws_size, hipStream_t stream) {
  const float* pos  = (const float*)d_in[0];
  const float* mass = (const float*)d_in[1];
  const float* W1   = (const float*)d_in[2];
  const float* b1   = (const float*)d_in[3];
  const float* W2   = (const float*)d_in[4];
  const float* b2   = (const float*)d_in[5];
  float* out = (float*)d_out;
  float* ws  = (float*)d_ws;   // [0,64): hraw, [64,128): h

  k_zero<<<dim3((KIN + 255) / 256), dim3(256), 0, stream>>>(out, ws);
  k_forces<<<dim3(NB / WG_I, JSLICES), dim3(256), 0, stream>>>(pos, mass, out);
  k_mlp1<<<dim3(24), dim3(256), 0, stream>>>(pos, W1, ws);
  k_mlp2<<<dim3(1), dim3(64), 0, stream>>>(ws, b1, ws + HDIM);
  k_mlp3<<<dim3(KIN / 256), dim3(256), 0, stream>>>(ws + HDIM, W2, b2, out);
}